// EmbedLoopyBP_128849018974
// MI455X (gfx1250) — compile-verified
//
#include <hip/hip_runtime.h>

// EmbedLoopyBP on MI455X (gfx1250, wave32).
// GEMMs use V_WMMA_F32_16X16X4_F32 (exact f32) with B-fragments pre-swizzled
// in LDS (one ds_load_b64 per fragment, conflict-free). Segment sums use
// global_atomic_add_f32. cur buffers ping-pong across BP iterations.

typedef __attribute__((ext_vector_type(2))) float v2f;
typedef __attribute__((ext_vector_type(8))) float v8f;

#define TPB 256              // 8 waves per block
#define WPB 8
#define ROWS_PER_BLOCK 128   // 16 rows per wave
#define SLDA 68              // LDS row stride (floats): stride-4 banks, conflict free

// ---- swizzled-W staging -----------------------------------------------------
// B-fragment layout for V_WMMA_F32_16X16X4_F32: for K-slab s (4 k's) and
// N-tile nt, lane holds {W[k0][col], W[k0+1][col]} with k0 = s*4 + (lane>>4)*2,
// col = nt*16 + (lane&15). Stored as float2 at [(s*4+nt)*32 + lane].
template <int K>
__device__ __forceinline__ void stage_W_swizzled(const float* __restrict__ W,
                                                 float2* __restrict__ sWsw, int tid) {
  constexpr int NFRAG = (K / 4) * 4 * 32;
  for (int d = tid; d < NFRAG; d += TPB) {
    int ln   = d & 31;
    int nt   = (d >> 5) & 3;
    int slab = d >> 7;
    int k0   = slab * 4 + (ln >> 4) * 2;
    int col  = nt * 16 + (ln & 15);
    float2 f;
    f.x = W[(size_t)k0 * 64 + col];
    f.y = W[(size_t)(k0 + 1) * 64 + col];
    sWsw[d] = f;
  }
}

// Wave-level 16xK @ Kx64 -> 16x64 f32 WMMA core.
// sA: 16 rows x SLDA floats (this wave's tile). sWsw: swizzled B fragments.
template <int K>
__device__ __forceinline__ void mma16xK(const float* __restrict__ sA,
                                        const float2* __restrict__ sWsw,
                                        v8f acc[4], int lane) {
  const int hl = lane >> 4;
  const int m  = lane & 15;
  #pragma unroll
  for (int kk = 0; kk < K; kk += 4) {
    const int slab = kk >> 2;
    v2f a = *(const v2f*)(sA + m * SLDA + kk + hl * 2);  // A fragment (1 ds_load_b64)
    #pragma unroll
    for (int nt = 0; nt < 4; ++nt) {
      v2f b = *(const v2f*)&sWsw[(slab * 4 + nt) * 32 + lane];
      acc[nt] = __builtin_amdgcn_wmma_f32_16x16x4_f32(false, a, false, b,
                                                      (short)0, acc[nt], false, false);
    }
  }
}

// ---------------- utility kernels ----------------
__global__ void k_zero(float* __restrict__ p, long long n) {
  long long i = (long long)blockIdx.x * blockDim.x + threadIdx.x;
  if (i < n) p[i] = 0.f;
}

__global__ void k_final_relu(const float* __restrict__ x, float* __restrict__ y, int n) {
  int i = blockIdx.x * blockDim.x + threadIdx.x;
  if (i < n) y[i] = fmaxf(x[i], 0.f);
}

// segment_sum(cur, dst) -> agg  (one atomic f32 add per element)
__global__ void k_scatter_add(const float* __restrict__ cur, const int* __restrict__ dst,
                              float* __restrict__ agg, int E) {
  long long i = (long long)blockIdx.x * blockDim.x + threadIdx.x;
  if (i >= (long long)E * 64) return;
  int e = (int)(i >> 6);
  int c = (int)(i & 63);
  atomicAdd(&agg[(size_t)dst[e] * 64 + c], cur[i]);
}

// ---------------- GEMM kernels ----------------

// Y[row] = X[row] @ W (K=64) + bias
__global__ void k_node_lin(const float* __restrict__ X, const float* __restrict__ W,
                           const float* __restrict__ bias, float* __restrict__ Y,
                           int nrows) {
  __shared__ __align__(16) float2 sW[16 * 4 * 32];
  __shared__ __align__(16) float  sA[WPB][16 * SLDA];
  const int tid = threadIdx.x;
  stage_W_swizzled<64>(W, sW, tid);
  const int wave = tid >> 5, lane = tid & 31;
  const int tile = (blockIdx.x * WPB + wave) * 16;
  {
    int r = lane >> 1, p = lane & 1;
    int row = tile + r;
    float* a = &sA[wave][r * SLDA];
    if (row < nrows) {
      const float* x = X + (size_t)row * 64;
      #pragma unroll
      for (int j = 0; j < 8; ++j) {
        int c = p * 32 + j * 4;
        float4 v = *(const float4*)(x + c);
        a[c] = v.x; a[c + 1] = v.y; a[c + 2] = v.z; a[c + 3] = v.w;
      }
    } else {
      #pragma unroll
      for (int j = 0; j < 8; ++j) {
        int c = p * 32 + j * 4;
        a[c] = a[c + 1] = a[c + 2] = a[c + 3] = 0.f;
      }
    }
  }
  __syncthreads();
  v8f acc[4] = {};
  mma16xK<64>(sA[wave], sW, acc, lane);
  const int hl = lane >> 4, m = lane & 15;
  #pragma unroll
  for (int nt = 0; nt < 4; ++nt) {
    const int col = nt * 16 + m;
    const float bc = bias[col];
    #pragma unroll
    for (int r = 0; r < 8; ++r) {
      int row = tile + hl * 8 + r;
      if (row < nrows) Y[(size_t)row * 64 + col] = acc[nt][r] + bc;
    }
  }
}

// msg = edge_feat @ We + be + nlin[src];  cur = relu(msg)   (K=32)
__global__ void k_edge_init(const float* __restrict__ X, const float* __restrict__ W,
                            const float* __restrict__ bias, const float* __restrict__ nlin,
                            const int* __restrict__ src, float* __restrict__ msg,
                            float* __restrict__ cur, int E) {
  __shared__ __align__(16) float2 sW[8 * 4 * 32];
  __shared__ __align__(16) float  sA[WPB][16 * SLDA];
  const int tid = threadIdx.x;
  stage_W_swizzled<32>(W, sW, tid);
  const int wave = tid >> 5, lane = tid & 31;
  const int tile = (blockIdx.x * WPB + wave) * 16;
  {
    int r = lane >> 1, p = lane & 1;
    int e = tile + r;
    float* a = &sA[wave][r * SLDA];
    if (e < E) {
      // prefetch the gathered nlin row used in the epilogue
      __builtin_prefetch(nlin + (size_t)src[e] * 64 + p * 32, 0, 0);
      const float* x = X + (size_t)e * 32;
      #pragma unroll
      for (int j = 0; j < 4; ++j) {
        int c = p * 16 + j * 4;
        float4 v = *(const float4*)(x + c);
        a[c] = v.x; a[c + 1] = v.y; a[c + 2] = v.z; a[c + 3] = v.w;
      }
    } else {
      #pragma unroll
      for (int j = 0; j < 4; ++j) {
        int c = p * 16 + j * 4;
        a[c] = a[c + 1] = a[c + 2] = a[c + 3] = 0.f;
      }
    }
  }
  __syncthreads();
  v8f acc[4] = {};
  mma16xK<32>(sA[wave], sW, acc, lane);
  const int hl = lane >> 4, m = lane & 15;
  #pragma unroll
  for (int nt = 0; nt < 4; ++nt) {
    const int col = nt * 16 + m;
    const float bc = bias[col];
    #pragma unroll
    for (int r = 0; r < 8; ++r) {
      int e = tile + hl * 8 + r;
      if (e < E) {
        float mv = acc[nt][r] + bc + nlin[(size_t)src[e] * 64 + col];
        msg[(size_t)e * 64 + col] = mv;
        cur[(size_t)e * 64 + col] = fmaxf(mv, 0.f);
      }
    }
  }
}

// curOut = relu( (nagg[src] - curIn[rev]) @ Wc + bc + msg )   (K=64)
__global__ void k_edge_update(const float* __restrict__ nagg, const float* __restrict__ curIn,
                              const int* __restrict__ src, const int* __restrict__ rev,
                              const float* __restrict__ msg, const float* __restrict__ W,
                              const float* __restrict__ bias, float* __restrict__ curOut,
                              int E) {
  __shared__ __align__(16) float2 sW[16 * 4 * 32];
  __shared__ __align__(16) float  sA[WPB][16 * SLDA];
  const int tid = threadIdx.x;
  const int wave = tid >> 5, lane = tid & 31;
  const int tile = (blockIdx.x * WPB + wave) * 16;
  {
    // prefetch the msg row used in the epilogue (addresses known up-front)
    int r = lane >> 1, p = lane & 1;
    int e = tile + r;
    if (e < E) __builtin_prefetch(msg + (size_t)e * 64 + p * 32, 0, 0);
  }
  stage_W_swizzled<64>(W, sW, tid);
  {
    int r = lane >> 1, p = lane & 1;
    int e = tile + r;
    float* a = &sA[wave][r * SLDA];
    if (e < E) {
      const float* ns = nagg  + (size_t)src[e] * 64;
      const float* cr = curIn + (size_t)rev[e] * 64;
      #pragma unroll
      for (int j = 0; j < 8; ++j) {
        int c = p * 32 + j * 4;
        float4 x = *(const float4*)(ns + c);
        float4 y = *(const float4*)(cr + c);
        a[c]     = x.x - y.x;
        a[c + 1] = x.y - y.y;
        a[c + 2] = x.z - y.z;
        a[c + 3] = x.w - y.w;
      }
    } else {
      #pragma unroll
      for (int j = 0; j < 8; ++j) {
        int c = p * 32 + j * 4;
        a[c] = a[c + 1] = a[c + 2] = a[c + 3] = 0.f;
      }
    }
  }
  __syncthreads();
  v8f acc[4] = {};
  mma16xK<64>(sA[wave], sW, acc, lane);
  const int hl = lane >> 4, m = lane & 15;
  #pragma unroll
  for (int nt = 0; nt < 4; ++nt) {
    const int col = nt * 16 + m;
    const float bc = bias[col];
    #pragma unroll
    for (int r = 0; r < 8; ++r) {
      int e = tile + hl * 8 + r;
      if (e < E) {
        float v = acc[nt][r] + bc + msg[(size_t)e * 64 + col];
        curOut[(size_t)e * 64 + col] = fmaxf(v, 0.f);
      }
    }
  }
}

// y = relu( relu(nagg) @ Wo + bo ); atomic add into gsum[graph_ids[node]]
__global__ void k_node_out(const float* __restrict__ nagg, const float* __restrict__ W,
                           const float* __restrict__ bias, const int* __restrict__ gid,
                           float* __restrict__ gsum, int nrows) {
  __shared__ __align__(16) float2 sW[16 * 4 * 32];
  __shared__ __align__(16) float  sA[WPB][16 * SLDA];
  const int tid = threadIdx.x;
  stage_W_swizzled<64>(W, sW, tid);
  const int wave = tid >> 5, lane = tid & 31;
  const int tile = (blockIdx.x * WPB + wave) * 16;
  {
    int r = lane >> 1, p = lane & 1;
    int row = tile + r;
    float* a = &sA[wave][r * SLDA];
    if (row < nrows) {
      const float* x = nagg + (size_t)row * 64;
      #pragma unroll
      for (int j = 0; j < 8; ++j) {
        int c = p * 32 + j * 4;
        float4 v = *(const float4*)(x + c);
        a[c]     = fmaxf(v.x, 0.f);
        a[c + 1] = fmaxf(v.y, 0.f);
        a[c + 2] = fmaxf(v.z, 0.f);
        a[c + 3] = fmaxf(v.w, 0.f);
      }
    } else {
      #pragma unroll
      for (int j = 0; j < 8; ++j) {
        int c = p * 32 + j * 4;
        a[c] = a[c + 1] = a[c + 2] = a[c + 3] = 0.f;
      }
    }
  }
  __syncthreads();
  v8f acc[4] = {};
  mma16xK<64>(sA[wave], sW, acc, lane);
  const int hl = lane >> 4, m = lane & 15;
  #pragma unroll
  for (int nt = 0; nt < 4; ++nt) {
    const int col = nt * 16 + m;
    const float bc = bias[col];
    #pragma unroll
    for (int r = 0; r < 8; ++r) {
      int row = tile + hl * 8 + r;
      if (row < nrows) {
        float v = fmaxf(acc[nt][r] + bc, 0.f);
        atomicAdd(&gsum[(size_t)gid[row] * 64 + col], v);
      }
    }
  }
}

// ---------------- host orchestration ----------------
static inline unsigned cdiv(long long a, long long b) { return (unsigned)((a + b - 1) / b); }

extern "C" void kernel_launch(void* const* d_in, const int* in_sizes, int n_in,
                              void* d_out, int out_size, void* d_ws, size_t ws_size,
                              hipStream_t stream) {
  const float* node_feat = (const float*)d_in[0];
  const float* edge_feat = (const float*)d_in[1];
  const int*   src       = (const int*)d_in[2];
  const int*   dst       = (const int*)d_in[3];
  const int*   rev       = (const int*)d_in[4];
  const int*   gid       = (const int*)d_in[5];
  // d_in[6] = n_graphs scalar (G derived from out_size instead)
  const float* Wn = (const float*)d_in[7];
  const float* bn = (const float*)d_in[8];
  const float* We = (const float*)d_in[9];
  const float* be = (const float*)d_in[10];
  const float* Wc = (const float*)d_in[11];
  const float* bc = (const float*)d_in[12];
  const float* Wo = (const float*)d_in[13];
  const float* bo = (const float*)d_in[14];

  const int N = in_sizes[0] / 64;   // 50000
  const int E = in_sizes[1] / 32;   // 800000 directed edges
  const int G = out_size / 64;      // 128

  // workspace carve-up (floats)
  float* ws   = (float*)d_ws;
  float* msg  = ws;                            // E*64
  float* curA = msg  + (size_t)E * 64;         // E*64
  float* curB = curA + (size_t)E * 64;         // E*64
  float* nlin = curB + (size_t)E * 64;         // N*64 (reused as nagg after init)
  float* nagg = nlin;                          // alias: nlin dead after k_edge_init
  float* gsum = nlin + (size_t)N * 64;         // G*64

  const unsigned gNode = cdiv(N, ROWS_PER_BLOCK);
  const unsigned gEdge = cdiv(E, ROWS_PER_BLOCK);
  const unsigned gScat = cdiv((long long)E * 64, TPB);
  const unsigned gZn   = cdiv((long long)N * 64, TPB);
  const unsigned gZg   = cdiv((long long)G * 64, TPB);

  // 1. node linear
  k_node_lin<<<gNode, TPB, 0, stream>>>(node_feat, Wn, bn, nlin, N);
  // 2. edge linear + gather + relu
  k_edge_init<<<gEdge, TPB, 0, stream>>>(edge_feat, We, be, nlin, src, msg, curA, E);

  // 3. loopy BP iterations (ping-pong cur buffers)
  float* cin = curA;
  float* cout = curB;
  for (int it = 0; it < 3; ++it) {
    k_zero<<<gZn, TPB, 0, stream>>>(nagg, (long long)N * 64);
    k_scatter_add<<<gScat, TPB, 0, stream>>>(cin, dst, nagg, E);
    k_edge_update<<<gEdge, TPB, 0, stream>>>(nagg, cin, src, rev, msg, Wc, bc, cout, E);
    float* t = cin; cin = cout; cout = t;
  }

  // 4. e2n pool, output linear, per-graph pool, final relu
  k_zero<<<gZn, TPB, 0, stream>>>(nagg, (long long)N * 64);
  k_scatter_add<<<gScat, TPB, 0, stream>>>(cin, dst, nagg, E);
  k_zero<<<gZg, TPB, 0, stream>>>(gsum, (long long)G * 64);
  k_node_out<<<gNode, TPB, 0, stream>>>(nagg, Wo, bo, gid, gsum, N);
  k_final_relu<<<gZg, TPB, 0, stream>>>(gsum, (float*)d_out, G * 64);
}